// ImageModel_50663434223985
// MI455X (gfx1250) — compile-verified
//
#include <hip/hip_runtime.h>

// ---------------------------------------------------------------------------
// Gaussian splat (N peaks -> 2048x2048 image) reformulated as tiled GATHER:
//   out[ty*16+m, tx*16+n] = bg + sum_k  h_k * gy_k(m) * gx_k(n)
// which per 16x16 tile is  C += A(16xK) x B(Kx16)  ->  V_WMMA_F32_16X16X4_F32
// (K = 4 peaks per WMMA, f32 end-to-end). Avoids 44M global float atomics of
// the scatter formulation; image is written exactly once (16MB, L2 resident).
// This revision: branch-free main batch loop (tail handled once, uniformly)
// so the hot loop is 4x s_load_b32 + 4x s_load_b128 + factor VALU + 1 WMMA.
// ---------------------------------------------------------------------------

#define IMG_W     2048
#define IMG_H     2048
#define HALF_WIN  10               // 21-wide window -> +-10 integer offsets
#define TILE      16
#define TPD       (IMG_W / TILE)   // 128 tiles per dim
#define NTILES    (TPD * TPD)      // 16384

typedef float v2f __attribute__((ext_vector_type(2)));
typedef float v8f __attribute__((ext_vector_type(8)));

__device__ __forceinline__ float gfac(float d, float inv2w2) {
  // Gaussian factor at (exact) integer offset d, masked to the 21-wide window.
  float f = __expf(-d * d * inv2w2);
  return (fabsf(d) <= (float)HALF_WIN) ? f : 0.0f;
}

// --- pass 1: per-peak params + tile histogram ------------------------------
__global__ void prep_kernel(const float* __restrict__ px, const float* __restrict__ py,
                            const float* __restrict__ ht, const float* __restrict__ wd,
                            float4* __restrict__ params, int* __restrict__ counts, int n) {
  int i = blockIdx.x * blockDim.x + threadIdx.x;
  if (i >= n) return;
  int xi = (int)rintf(px[i]);           // round-to-nearest-even == jnp.round
  int yi = (int)rintf(py[i]);
  xi = min(max(xi, 0), IMG_W - 1);
  yi = min(max(yi, 0), IMG_H - 1);
  float w = wd[i];
  params[i] = make_float4((float)xi, (float)yi, ht[i], 0.5f / (w * w));
  int tx0 = max(xi - HALF_WIN, 0) >> 4, tx1 = min(xi + HALF_WIN, IMG_W - 1) >> 4;
  int ty0 = max(yi - HALF_WIN, 0) >> 4, ty1 = min(yi + HALF_WIN, IMG_H - 1) >> 4;
  for (int t = ty0; t <= ty1; ++t)
    for (int s = tx0; s <= tx1; ++s)
      atomicAdd(&counts[t * TPD + s], 1);
}

// --- pass 2: exclusive prefix sum over 16384 tile counts (one block) -------
__global__ void __launch_bounds__(1024) scan_kernel(const int* __restrict__ counts,
                                                    int* __restrict__ offsets) {
  __shared__ int sh[1024];
  const int tid = threadIdx.x;
  int local[16];
  int s = 0;
#pragma unroll
  for (int i = 0; i < 16; ++i) { local[i] = s; s += counts[tid * 16 + i]; }
  sh[tid] = s;
  __syncthreads();
  for (int off = 1; off < 1024; off <<= 1) {
    int v = (tid >= off) ? sh[tid - off] : 0;
    __syncthreads();
    sh[tid] += v;
    __syncthreads();
  }
  const int tbase = sh[tid] - s;  // exclusive
#pragma unroll
  for (int i = 0; i < 16; ++i) offsets[tid * 16 + i] = tbase + local[i];
}

// --- pass 3: fill per-tile peak-index lists --------------------------------
__global__ void fill_kernel(const float* __restrict__ px, const float* __restrict__ py,
                            const int* __restrict__ offsets, int* __restrict__ cursor,
                            int* __restrict__ list, int n) {
  int i = blockIdx.x * blockDim.x + threadIdx.x;
  if (i >= n) return;
  int xi = (int)rintf(px[i]);
  int yi = (int)rintf(py[i]);
  xi = min(max(xi, 0), IMG_W - 1);
  yi = min(max(yi, 0), IMG_H - 1);
  int tx0 = max(xi - HALF_WIN, 0) >> 4, tx1 = min(xi + HALF_WIN, IMG_W - 1) >> 4;
  int ty0 = max(yi - HALF_WIN, 0) >> 4, ty1 = min(yi + HALF_WIN, IMG_H - 1) >> 4;
  for (int t = ty0; t <= ty1; ++t)
    for (int s = tx0; s <= tx1; ++s) {
      int tl = t * TPD + s;
      int slot = offsets[tl] + atomicAdd(&cursor[tl], 1);
      list[slot] = i;
    }
}

// --- pass 4: one wave32 per 16x16 tile, WMMA-accumulated gather ------------
__global__ void __launch_bounds__(256) gather_kernel(
    const float4* __restrict__ params, const int* __restrict__ counts,
    const int* __restrict__ offsets, const int* __restrict__ list,
    const float* __restrict__ bg_p, float* __restrict__ out) {
  const int lane = threadIdx.x & 31;
  const int tile = blockIdx.x * (blockDim.x >> 5) + (threadIdx.x >> 5);
  const int tx = (tile % TPD) * TILE;
  const int ty = (tile / TPD) * TILE;
  const int half = lane >> 4;    // lane group 0-15 vs 16-31
  const int l = lane & 15;
  const float fx = (float)(tx + l);    // this lane's column coordinate (B / N)
  const float fy = (float)(ty + l);    // this lane's row coordinate    (A / M)
  const float bg = bg_p[0];

  // wave-uniform scalars: keeps all control flow non-divergent so EXEC stays
  // all-ones across the WMMA (ISA requirement) and list/params loads lower
  // to s_load_b32 / s_load_b128 broadcasts.
  const int cnt  = __builtin_amdgcn_readfirstlane(counts[tile]);
  const int base = __builtin_amdgcn_readfirstlane(offsets[tile]);
  const int nb   = cnt & ~3;     // full 4-peak batches

  v8f acc = {};

  // -- branch-free main loop: exactly 4 peaks per iteration -----------------
#pragma unroll 2
  for (int b = 0; b < nb; b += 4) {
    float4 P[4];
#pragma unroll
    for (int k = 0; k < 4; ++k) {
      const int pidx = __builtin_amdgcn_readfirstlane(list[base + b + k]);
      P[k] = params[pidx];
    }
    // 32-bit A (16x4) layout: lanes 0-15 hold {K0,K1} in VGPR0/1, lanes 16-31
    // hold {K2,K3}. B (4x16) rows striped across lanes per C/D convention.
    const float pax = half ? P[2].x : P[0].x;
    const float pay = half ? P[2].y : P[0].y;
    const float pah = half ? P[2].z : P[0].z;
    const float paw = half ? P[2].w : P[0].w;
    const float pbx = half ? P[3].x : P[1].x;
    const float pby = half ? P[3].y : P[1].y;
    const float pbh = half ? P[3].z : P[1].z;
    const float pbw = half ? P[3].w : P[1].w;

    v2f A, B;
    A.x = gfac(fy - pay, paw) * pah;   // A[m][k] = h_k * gy_k(m)
    A.y = gfac(fy - pby, pbw) * pbh;
    B.x = gfac(fx - pax, paw);         // B[k][n] = gx_k(n)
    B.y = gfac(fx - pbx, pbw);

    // C += A(16x4) x B(4x16): 4 peaks' rank-1 patches in one instruction.
    acc = __builtin_amdgcn_wmma_f32_16x16x4_f32(
        false, A, false, B, (short)0, acc, false, false);
  }

  // -- uniform remainder batch (1..3 peaks), executed at most once ----------
  if (nb < cnt) {
    float4 P[4];
#pragma unroll
    for (int k = 0; k < 4; ++k) {
      const int li = nb + k;                       // wave-uniform
      if (li < cnt) {
        const int pidx = __builtin_amdgcn_readfirstlane(list[base + li]);
        P[k] = params[pidx];
      } else {
        // far-away zero-height dummy peak -> factors exactly 0
        P[k] = make_float4(1.0e9f, 1.0e9f, 0.0f, 1.0f);
      }
    }
    const float pax = half ? P[2].x : P[0].x;
    const float pay = half ? P[2].y : P[0].y;
    const float pah = half ? P[2].z : P[0].z;
    const float paw = half ? P[2].w : P[0].w;
    const float pbx = half ? P[3].x : P[1].x;
    const float pby = half ? P[3].y : P[1].y;
    const float pbh = half ? P[3].z : P[1].z;
    const float pbw = half ? P[3].w : P[1].w;

    v2f A, B;
    A.x = gfac(fy - pay, paw) * pah;
    A.y = gfac(fy - pby, pbw) * pbh;
    B.x = gfac(fx - pax, paw);
    B.y = gfac(fx - pbx, pbw);

    acc = __builtin_amdgcn_wmma_f32_16x16x4_f32(
        false, A, false, B, (short)0, acc, false, false);
  }

  // C/D layout: VGPR r -> (M = r + 8*half, N = l). Single coalesced write of
  // the whole image (16 MB) -- the only HBM-heavy traffic in the pipeline.
#pragma unroll
  for (int r = 0; r < 8; ++r) {
    const int m = r + 8 * half;
    out[(ty + m) * IMG_W + (tx + l)] = bg + acc[r];
  }
}

// ---------------------------------------------------------------------------
extern "C" void kernel_launch(void* const* d_in, const int* in_sizes, int n_in,
                              void* d_out, int out_size, void* d_ws, size_t ws_size,
                              hipStream_t stream) {
  // setup_inputs order: X, Y, pos_x, pos_y, height, width, background
  const float* pos_x  = (const float*)d_in[2];
  const float* pos_y  = (const float*)d_in[3];
  const float* height = (const float*)d_in[4];
  const float* width  = (const float*)d_in[5];
  const float* bg     = (const float*)d_in[6];
  float* out = (float*)d_out;
  const int n = in_sizes[2];

  // workspace layout: [counts NT][cursor NT][offsets NT][params N*f4][list 9N]
  char* w = (char*)d_ws;
  int*    counts  = (int*)(w);
  int*    cursor  = (int*)(w + (size_t)NTILES * 4);
  int*    offsets = (int*)(w + (size_t)NTILES * 8);
  float4* params  = (float4*)(w + (size_t)NTILES * 12);
  int*    list    = (int*)(w + (size_t)NTILES * 12 + (size_t)n * 16);

  hipMemsetAsync(counts, 0, (size_t)NTILES * 8, stream);  // counts + cursor

  const int tpb = 256;
  prep_kernel<<<(n + tpb - 1) / tpb, tpb, 0, stream>>>(pos_x, pos_y, height, width,
                                                       params, counts, n);
  scan_kernel<<<1, 1024, 0, stream>>>(counts, offsets);
  fill_kernel<<<(n + tpb - 1) / tpb, tpb, 0, stream>>>(pos_x, pos_y, offsets, cursor,
                                                       list, n);
  gather_kernel<<<NTILES / 8, 256, 0, stream>>>(params, counts, offsets, list, bg, out);
}